// GCN_2_layers_10376640987637
// MI455X (gfx1250) — compile-verified
//
#include <hip/hip_runtime.h>
#include <hip/hip_bf16.h>

typedef float v2f __attribute__((ext_vector_type(2)));
typedef float v8f __attribute__((ext_vector_type(8)));

#define F 64

// ---------------- degree / normalization ----------------

__global__ void gcn_init_deg(float* __restrict__ deg, int N) {
  int i = blockIdx.x * blockDim.x + threadIdx.x;
  if (i < N) deg[i] = 1.0f;  // self loop contributes 1 to every node's degree
}

__global__ void gcn_count_deg(const long long* __restrict__ dst, float* __restrict__ deg, int E) {
  int e = blockIdx.x * blockDim.x + threadIdx.x;
  if (e < E) atomicAdd(&deg[(int)dst[e]], 1.0f);
}

__global__ void gcn_rsqrt_deg(float* __restrict__ deg, int N) {
  int i = blockIdx.x * blockDim.x + threadIdx.x;
  if (i < N) { float d = deg[i]; deg[i] = d > 0.0f ? rsqrtf(d) : 0.0f; }
}

// ---------------- GEMM: H = (relu?)A[N,64] x W[64,64] via V_WMMA_F32_16X16X4_F32 ----

// Wave-level 16x16 output tile, K looped in steps of 4 (16 chained WMMAs).
// A fragment (16x4 f32, 2 VGPRs): lane<16 -> K=k..k+1 of row (lane&15),
//                                 lane>=16 -> K=k+2..k+3 of same row.
// B fragment mirrors A from row-major W (two strided scalar loads).
// C/D: 8 VGPRs, lane<16 -> rows 0..7, lane>=16 -> rows 8..15, col = lane&15.
template <bool RELU_IN>
__global__ void gcn_gemm64_wmma(const float* __restrict__ A, const float* __restrict__ W,
                                float* __restrict__ H) {
  const int lane = threadIdx.x & 31;
  const int wave = threadIdx.x >> 5;
  const int r0 = blockIdx.x * 32 + (wave >> 2) * 16;  // 8 waves: 2 row tiles x 4 col tiles
  const int c0 = (wave & 3) * 16;
  const int m = lane & 15;
  const int half = lane >> 4;

  const float* arow = A + (size_t)(r0 + m) * F;
  v8f acc = {};
#pragma unroll
  for (int k = 0; k < F; k += 4) {
    const int kk = k + 2 * half;
    v2f a = *(const v2f*)(arow + kk);
    if (RELU_IN) { a.x = fmaxf(a.x, 0.0f); a.y = fmaxf(a.y, 0.0f); }
    v2f b;
    b.x = W[(size_t)kk * F + c0 + m];
    b.y = W[(size_t)(kk + 1) * F + c0 + m];
    acc = __builtin_amdgcn_wmma_f32_16x16x4_f32(
        /*neg_a=*/false, a, /*neg_b=*/false, b,
        /*c_mod=*/(short)0, acc, /*reuse_a=*/false, /*reuse_b=*/false);
  }

  const int mb = half * 8;
#pragma unroll
  for (int r = 0; r < 8; ++r)
    H[(size_t)(r0 + mb + r) * F + c0 + m] = acc[r];
}

// Scalar tail for rows not covered by the 32-row WMMA grid (unused when N%32==0).
template <bool RELU_IN>
__global__ void gcn_gemm64_tail(const float* __restrict__ A, const float* __restrict__ W,
                                float* __restrict__ H, int n0, int N) {
  int t = blockIdx.x * blockDim.x + threadIdx.x;
  int row = n0 + t / F, col = t % F;
  if (row >= N) return;
  float s = 0.0f;
  for (int k = 0; k < F; ++k) {
    float a = A[(size_t)row * F + k];
    if (RELU_IN) a = fmaxf(a, 0.0f);
    s += a * W[(size_t)k * F + col];
  }
  H[(size_t)row * F + col] = s;
}

// ---------------- aggregation ----------------

// agg[node] = bias + dis[node]^2 * h[node]   (self-loop message + bias, fused init)
__global__ void gcn_self_bias(const float* __restrict__ h, const float* __restrict__ dis,
                              const float* __restrict__ bias, float* __restrict__ agg, int N) {
  int t = blockIdx.x * blockDim.x + threadIdx.x;
  int node = t >> 5;
  int f = (t & 31) * 2;
  if (node >= N) return;
  float w = dis[node];
  w *= w;
  v2f hv = *(const v2f*)(h + (size_t)node * F + f);
  v2f o;
  o.x = bias[f] + w * hv.x;
  o.y = bias[f + 1] + w * hv.y;
  *(v2f*)(agg + (size_t)node * F + f) = o;
}

// One wave32 per edge: 32 lanes x float2 = one 64-feature row; L2-resident atomics.
__global__ void gcn_scatter(const long long* __restrict__ src, const long long* __restrict__ dst,
                            const float* __restrict__ dis, const float* __restrict__ h,
                            float* __restrict__ agg, int E) {
  int edge = blockIdx.x * (blockDim.x >> 5) + (threadIdx.x >> 5);
  if (edge >= E) return;
  int lane = threadIdx.x & 31;
  int s = (int)src[edge];
  int d = (int)dst[edge];
  float nrm = dis[s] * dis[d];
  v2f hv = *(const v2f*)(h + (size_t)s * F + lane * 2);
  atomicAdd(agg + (size_t)d * F + lane * 2, nrm * hv.x);
  atomicAdd(agg + (size_t)d * F + lane * 2 + 1, nrm * hv.y);
}

// ---------------- driver ----------------

extern "C" void kernel_launch(void* const* d_in, const int* in_sizes, int n_in,
                              void* d_out, int out_size, void* d_ws, size_t ws_size,
                              hipStream_t stream) {
  const float* x = (const float*)d_in[0];
  const long long* ei = (const long long*)d_in[1];  // int64 [2, E]
  const float* W1 = (const float*)d_in[2];
  const float* b1 = (const float*)d_in[3];
  const float* W2 = (const float*)d_in[4];
  const float* b2 = (const float*)d_in[5];
  float* out = (float*)d_out;

  const int N = in_sizes[0] / F;
  const int E = in_sizes[1] / 2;
  const long long* srcI = ei;
  const long long* dstI = ei + E;

  // workspace: dis[N] | h[N*64] | agg[N*64]  (~51.6 MB, all fits in 192 MB L2)
  float* dis = (float*)d_ws;
  size_t nAl = ((size_t)N + 63) & ~(size_t)63;  // 256B-align next region
  float* h = dis + nAl;
  float* agg = h + (size_t)N * F;

  const int T = 256;
  const int nWmma = N & ~31;
  const int rem = N - nWmma;
  const int selfBlocks = (int)(((size_t)N * 32 + T - 1) / T);
  const int edgeBlocks = (E + 7) / 8;  // 8 waves/block, 1 edge/wave

  // degree -> dis = deg^{-1/2}
  gcn_init_deg<<<(N + T - 1) / T, T, 0, stream>>>(dis, N);
  gcn_count_deg<<<(E + T - 1) / T, T, 0, stream>>>(dstI, dis, E);
  gcn_rsqrt_deg<<<(N + T - 1) / T, T, 0, stream>>>(dis, N);

  // layer 1: h = x @ W1 ; agg = b1 + dis^2*h + scatter(norm*h[src])
  if (nWmma) gcn_gemm64_wmma<false><<<nWmma / 32, 256, 0, stream>>>(x, W1, h);
  if (rem) gcn_gemm64_tail<false><<<(rem * F + T - 1) / T, T, 0, stream>>>(x, W1, h, nWmma, N);
  gcn_self_bias<<<selfBlocks, T, 0, stream>>>(h, dis, b1, agg, N);
  gcn_scatter<<<edgeBlocks, 256, 0, stream>>>(srcI, dstI, dis, h, agg, E);

  // layer 2: h = relu(agg) @ W2 ; out = b2 + dis^2*h + scatter(norm*h[src])
  if (nWmma) gcn_gemm64_wmma<true><<<nWmma / 32, 256, 0, stream>>>(agg, W2, h);
  if (rem) gcn_gemm64_tail<true><<<(rem * F + T - 1) / T, T, 0, stream>>>(agg, W2, h, nWmma, N);
  gcn_self_bias<<<selfBlocks, T, 0, stream>>>(h, dis, b2, out, N);
  gcn_scatter<<<edgeBlocks, 256, 0, stream>>>(srcI, dstI, dis, h, out, E);
}